// EGATConv_54159537602883
// MI455X (gfx1250) — compile-verified
//
#include <hip/hip_runtime.h>
#include <hip/hip_bf16.h>

// ---------------------------------------------------------------------------
// EGATConv for MI455X (gfx1250): fp32 WMMA (V_WMMA_F32_16X16X4_F32) for both
// GEMMs, atomics-based segment softmax + aggregation.
// ---------------------------------------------------------------------------

#define NN   50000
#define EE   800000
#define DIN  128
#define FIN  7
#define HH   4
#define DN   64
#define FE   32
#define KSTACK 263            // FIN + 2*DIN
#define NE_COL 128            // H*FE
#define HD_COL 256            // H*DN
#define NEG_SLOPE 0.01f

typedef __attribute__((ext_vector_type(2))) float v2f;
typedef __attribute__((ext_vector_type(8))) float v8f;

static __device__ __forceinline__ v8f v8f_zero() {
    v8f z = {0.f, 0.f, 0.f, 0.f, 0.f, 0.f, 0.f, 0.f};
    return z;
}

// One WMMA K-step against 8 column tiles of a [K,128] B matrix.
static __device__ __forceinline__ void wmma_step8(
    v8f acc[8], float ax, float ay,
    const float* __restrict__ B, int brow0, int brow1, int r) {
    v2f a; a.x = ax; a.y = ay;
#pragma unroll
    for (int t = 0; t < 8; ++t) {
        v2f b;
        b.x = B[brow0 * NE_COL + t * 16 + r];
        b.y = B[brow1 * NE_COL + t * 16 + r];
        acc[t] = __builtin_amdgcn_wmma_f32_16x16x4_f32(
            false, a, false, b, (short)0, acc[t], false, false);
    }
}

// ---------------------------------------------------------------------------
// Kernel 0: init accumulators (h_out region of d_out, amax, denom)
// ---------------------------------------------------------------------------
__global__ void k_init(float* __restrict__ h_out,
                       float* __restrict__ amax,
                       float* __restrict__ denom,
                       int n_out, int n_seg) {
    int i = blockIdx.x * blockDim.x + threadIdx.x;
    if (i < n_out) h_out[i] = 0.f;
    if (i < n_seg) {
        amax[i]  = -__builtin_inff();
        denom[i] = 0.f;
    }
}

// ---------------------------------------------------------------------------
// Kernel 1: node projection  h = nfeats @ W_nodes + b   [N,128]x[128,256]
// One wave per (16-row tile, 128-col half). grid = (3125, 2), block = 32.
// ---------------------------------------------------------------------------
__global__ void __launch_bounds__(32)
k_node_proj(const float* __restrict__ nfeats,   // [N,128]
            const float* __restrict__ Wn,       // [128,256]
            const float* __restrict__ bn,       // [256]
            float* __restrict__ h)              // [N,256]
{
    const int lane  = threadIdx.x;          // 0..31
    const int r     = lane & 15;            // row-in-tile / col-in-tile
    const int kHi   = (lane >> 4) * 2;      // 0 (lanes 0-15) or 2 (lanes 16-31)
    const int row   = blockIdx.x * 16 + r;  // N = 3125*16 exactly
    const int nBase = blockIdx.y * 128;

    v8f acc[8];
#pragma unroll
    for (int t = 0; t < 8; ++t) acc[t] = v8f_zero();

    const float* arow = nfeats + row * DIN;
#pragma unroll 4
    for (int ks = 0; ks < DIN / 4; ++ks) {
        const int ka = ks * 4;
        v2f a;
        a.x = arow[ka + kHi];
        a.y = arow[ka + kHi + 1];
#pragma unroll
        for (int t = 0; t < 8; ++t) {
            const int n = nBase + t * 16 + r;
            v2f b;
            b.x = Wn[(ka + kHi)     * HD_COL + n];
            b.y = Wn[(ka + kHi + 1) * HD_COL + n];
            acc[t] = __builtin_amdgcn_wmma_f32_16x16x4_f32(
                false, a, false, b, (short)0, acc[t], false, false);
        }
    }

    // D layout: VGPR i -> row (tile*16 + i + 8*(lane>=16)), col = nBase+t*16+r
    const int rOut = blockIdx.x * 16 + (lane >> 4) * 8;
#pragma unroll
    for (int t = 0; t < 8; ++t) {
        const int col  = nBase + t * 16 + r;
        const float bv = bn[col];
#pragma unroll
        for (int i = 0; i < 8; ++i)
            h[(rOut + i) * HD_COL + col] = acc[t][i] + bv;
    }
}

// ---------------------------------------------------------------------------
// Kernel 2: edge MLP  f_out = leaky_relu([h_src || f || h_dst] @ W_edges)
// Virtual A matrix [E, 263] gathered on the fly, K split into three STATIC
// phases (src rows / efeats boundary / dst rows) so no dynamic selects.
// One wave per 16-edge tile (8 accumulator tiles = full 128 output cols).
// block = 128 (4 waves), grid = 12500 -> 50000 tiles = E/16 exactly.
// ---------------------------------------------------------------------------
__global__ void __launch_bounds__(128)
k_edge_mlp(const float* __restrict__ nfeats,   // [N,128]
           const float* __restrict__ efeats,   // [E,7]
           const int*   __restrict__ src,
           const int*   __restrict__ dst,
           const float* __restrict__ We,       // [263,128]
           float* __restrict__ f_out)          // [E,128]
{
    const int lane = threadIdx.x & 31;
    const int tile = blockIdx.x * 4 + (threadIdx.x >> 5);  // 16-edge tile
    const int r    = lane & 15;
    const int kHi  = (lane >> 4) * 2;
    const int e    = tile * 16 + r;
    const int s    = src[e];
    const int d    = dst[e];

    __builtin_prefetch(We, 0, 3);   // gfx1250 global_prefetch_b8

    const float* srow = nfeats + (size_t)s * DIN;
    const float* drow = nfeats + (size_t)d * DIN;
    const float* erow = efeats + (size_t)e * FIN;

    v8f acc[8];
#pragma unroll
    for (int t = 0; t < 8; ++t) acc[t] = v8f_zero();

    // Phase 1: k in [0,128): A = nfeats[src]
#pragma unroll 4
    for (int ks = 0; ks < 32; ++ks) {
        const int k0 = ks * 4 + kHi;
        wmma_step8(acc, srow[k0], srow[k0 + 1], We, k0, k0 + 1, r);
    }
    // Phase 2a: k in [128,132): A = efeats[e][0..3]
    wmma_step8(acc, erow[kHi], erow[kHi + 1], We, 128 + kHi, 129 + kHi, r);
    // Phase 2b: k in [132,136): efeats[e][4..6] then nfeats[dst][0]
    {
        const float ax = (kHi == 0) ? erow[4] : erow[6];
        const float ay = (kHi == 0) ? erow[5] : drow[0];
        wmma_step8(acc, ax, ay, We, 132 + kHi, 133 + kHi, r);
    }
    // Phase 3: k in [136,260): A = nfeats[dst][1..124]
#pragma unroll 4
    for (int ks = 0; ks < 31; ++ks) {
        const int k0 = 136 + ks * 4 + kHi;          // B row
        wmma_step8(acc, drow[k0 - 135], drow[k0 - 134], We, k0, k0 + 1, r);
    }
    // Phase 3 tail: k in [260,264): dst[125..127], k=263 is zero padding
    {
        const float ax = drow[125 + kHi];                       // 125 or 127
        const float ay = (kHi == 0) ? drow[126] : 0.f;          // 126 or pad
        v2f a; a.x = ax; a.y = ay;
#pragma unroll
        for (int t = 0; t < 8; ++t) {
            v2f b;
            b.x = We[(260 + kHi) * NE_COL + t * 16 + r];        // row 260/262
            b.y = (kHi == 0) ? We[261 * NE_COL + t * 16 + r] : 0.f;
            acc[t] = __builtin_amdgcn_wmma_f32_16x16x4_f32(
                false, a, false, b, (short)0, acc[t], false, false);
        }
    }

    // epilogue: leaky ReLU, write f_out
    const int eOut = tile * 16 + (lane >> 4) * 8;
#pragma unroll
    for (int t = 0; t < 8; ++t) {
        const int col = t * 16 + r;
#pragma unroll
        for (int i = 0; i < 8; ++i) {
            float v = acc[t][i];
            v = (v > 0.f) ? v : NEG_SLOPE * v;
            f_out[(eOut + i) * NE_COL + col] = v;
        }
    }
}

// ---------------------------------------------------------------------------
// Kernel 3: attention scores a[e,h] = sum_j f_out[e,h,j] * (sum_k Wattn[k,j])
// plus per-destination atomic max. One thread per (e,h).
// ---------------------------------------------------------------------------
static __device__ __forceinline__ void atomic_max_f32(float* addr, float val) {
    unsigned int* ai = (unsigned int*)addr;
    unsigned int old = __hip_atomic_load(ai, __ATOMIC_RELAXED,
                                         __HIP_MEMORY_SCOPE_AGENT);
    while (true) {
        float cur = __uint_as_float(old);
        if (cur >= val) break;
        unsigned int prev = atomicCAS(ai, old, __float_as_uint(val));
        if (prev == old) break;
        old = prev;
    }
}

__global__ void __launch_bounds__(256)
k_attn_scores(const float* __restrict__ f_out,  // [E,128]
              const float* __restrict__ Wattn,  // [4,32]
              const int*   __restrict__ dst,
              float* __restrict__ a,            // [E,4]
              float* __restrict__ amax)         // [N,4]
{
    __shared__ float wsum[FE];
    if (threadIdx.x < FE) {
        const int j = threadIdx.x;
        wsum[j] = Wattn[j] + Wattn[FE + j] + Wattn[2 * FE + j] +
                  Wattn[3 * FE + j];
    }
    __syncthreads();

    const int idx = blockIdx.x * blockDim.x + threadIdx.x;  // E*4 exact
    const int e   = idx >> 2;
    const int hh  = idx & 3;
    const float4* f = (const float4*)(f_out + e * NE_COL + hh * FE);
    float sum = 0.f;
#pragma unroll
    for (int j4 = 0; j4 < FE / 4; ++j4) {
        const float4 v = f[j4];
        sum += v.x * wsum[j4 * 4 + 0] + v.y * wsum[j4 * 4 + 1] +
               v.z * wsum[j4 * 4 + 2] + v.w * wsum[j4 * 4 + 3];
    }
    a[idx] = sum;
    atomic_max_f32(&amax[dst[e] * HH + hh], sum);
}

// ---------------------------------------------------------------------------
// Kernel 4: ex = exp(a - amax[dst]) (in place), denom[dst] += ex
// ---------------------------------------------------------------------------
__global__ void __launch_bounds__(256)
k_softmax_denom(float* __restrict__ a,            // [E,4], overwritten by ex
                const float* __restrict__ amax,   // [N,4]
                const int*   __restrict__ dst,
                float* __restrict__ denom)        // [N,4]
{
    const int idx = blockIdx.x * blockDim.x + threadIdx.x;
    const int e   = idx >> 2;
    const int hh  = idx & 3;
    const float ev = __expf(a[idx] - amax[dst[e] * HH + hh]);
    a[idx] = ev;
    atomicAdd(&denom[dst[e] * HH + hh], ev);
}

// ---------------------------------------------------------------------------
// Kernel 5: h_out[dst] += (ex/denom[dst]) * h[src].  One block per edge.
// ---------------------------------------------------------------------------
__global__ void __launch_bounds__(256)
k_aggregate(const float* __restrict__ h,      // [N,256]
            const float* __restrict__ ex,     // [E,4]
            const float* __restrict__ denom,  // [N,4]
            const int*   __restrict__ src,
            const int*   __restrict__ dst,
            float* __restrict__ h_out)        // [N,256]
{
    const int e  = blockIdx.x;
    const int t  = threadIdx.x;        // 0..255 -> (head, dn)
    const int hh = t >> 6;
    const int s  = src[e];
    const int d  = dst[e];
    const float alpha = ex[e * HH + hh] / denom[d * HH + hh];
    atomicAdd(&h_out[d * HD_COL + t], alpha * h[s * HD_COL + t]);
}

// ---------------------------------------------------------------------------
// Launcher
// ---------------------------------------------------------------------------
extern "C" void kernel_launch(void* const* d_in, const int* in_sizes, int n_in,
                              void* d_out, int out_size, void* d_ws,
                              size_t ws_size, hipStream_t stream) {
    (void)in_sizes; (void)n_in; (void)out_size; (void)ws_size;

    const float* nfeats = (const float*)d_in[0];
    const float* efeats = (const float*)d_in[1];
    const int*   src    = (const int*)d_in[2];
    const int*   dst    = (const int*)d_in[3];
    const float* Wn     = (const float*)d_in[4];
    const float* bn     = (const float*)d_in[5];
    const float* We     = (const float*)d_in[6];
    const float* Wattn  = (const float*)d_in[7];

    // d_out = [h_out (N*H*DN) | f_out (E*H*FE)]
    float* h_out = (float*)d_out;
    float* f_out = (float*)d_out + (size_t)NN * HD_COL;

    // workspace layout (floats)
    float* ws    = (float*)d_ws;
    float* h     = ws;                                  // N*256 = 12.8M
    float* a_ex  = h + (size_t)NN * HD_COL;             // E*4   =  3.2M
    float* amax  = a_ex + (size_t)EE * HH;              // N*4
    float* denom = amax + (size_t)NN * HH;              // N*4

    const int n_out = NN * HD_COL;   // 12.8M
    const int n_seg = NN * HH;       // 200K

    // 0) init accumulators
    k_init<<<(n_out + 255) / 256, 256, 0, stream>>>(h_out, amax, denom,
                                                    n_out, n_seg);
    // 1) node projection (WMMA f32)
    k_node_proj<<<dim3(NN / 16, 2), 32, 0, stream>>>(nfeats, Wn, bn, h);
    // 2) edge MLP (WMMA f32) -> f_out part of d_out
    k_edge_mlp<<<EE / 16 / 4, 128, 0, stream>>>(nfeats, efeats, src, dst,
                                                We, f_out);
    // 3) attention scores + segment max
    k_attn_scores<<<EE * HH / 256, 256, 0, stream>>>(f_out, Wattn, dst,
                                                     a_ex, amax);
    // 4) exp + segment sum
    k_softmax_denom<<<EE * HH / 256, 256, 0, stream>>>(a_ex, amax, dst, denom);
    // 5) weighted aggregation
    k_aggregate<<<EE, 256, 0, stream>>>(h, a_ex, denom, src, dst, h_out);
}